// AttentionGCNLayer_28011776704812
// MI455X (gfx1250) — compile-verified
//
#include <hip/hip_runtime.h>

// ---------------------------------------------------------------------------
// AttentionGCNLayer for MI455X (gfx1250), fp32 end-to-end via
// V_WMMA_F32_16X16X4_F32 (exact fp32 matrix MACC, wave32).
// Compute-bound (~7e10 fp32 FLOP vs ~0.4GB traffic @ 23.3TB/s), so all
// GEMM-shaped work goes through the WMMA pipe with LDS-staged tiles.
// v2: transposed B tile in LDS (b64 frags, no packing movs), 32x32 wave tiles
//     (1 ds_load_b64 per WMMA), 2-m-tile scores kernel, next-chunk prefetch.
// ---------------------------------------------------------------------------

#define B_  4
#define N_  2048
#define D_  256
#define E_  3
#define H_  2
#define BN_ (B_ * N_)
#define NN_ ((long)N_ * N_)
#define ND_ ((long)N_ * D_)
#define NEGV (-1000000000.0f)

typedef __attribute__((ext_vector_type(2))) float v2f;
typedef __attribute__((ext_vector_type(8))) float v8f;

static __device__ __forceinline__ v8f wmma4(v2f a, v2f b, v8f c) {
  // D = A(16x4,f32) * B(4x16,f32) + C(16x16,f32), one wave.
  return __builtin_amdgcn_wmma_f32_16x16x4_f32(
      /*neg_a=*/false, a, /*neg_b=*/false, b,
      /*c_mod=*/(short)0, c, /*reuse_a=*/false, /*reuse_b=*/false);
}

// ---------------------------------------------------------------------------
// Generic fp32 WMMA GEMM:  C[M,Nc] = A[M,K] @ Bw[K,Nc]  (+bias / fused GCN)
//   block tile 128x64, 8 waves as 4(M)x2(N); wave tile 32x32 = 2x2 subtiles.
//   Inner step: 2 A-frag + 2 B-frag ds_load_b64 -> 4 WMMAs (full reuse).
//   mode 0: C = acc (+ bias[col] if bias)
//   mode 1: C = Xin[row,col] + relu(acc / degs[row] + bias[col])   (GCN step)
// Batched over blockIdx.z with explicit strides. Nc is gridDim.y*64.
// ---------------------------------------------------------------------------
__global__ __launch_bounds__(256) void gemm_wmma_f32(
    const float* __restrict__ A,  int lda, long aBS,
    const float* __restrict__ Bw, int ldb, long bBS,
    const float* __restrict__ bias,
    float* __restrict__ C, int ldc, long cBS,
    const float* __restrict__ Xin, int ldx, long xBS,
    const float* __restrict__ degs, long dBS,
    int K, int mode) {
  __shared__ float As[128][36];  // [m][k], +4 pad: conflict-free b64 frags
  __shared__ float Bt[64][36];   // [n][k] transposed: frag = contiguous b64

  const int bz = blockIdx.z;
  A  += (long)bz * aBS;
  Bw += (long)bz * bBS;
  C  += (long)bz * cBS;
  if (Xin)  Xin  += (long)bz * xBS;
  if (degs) degs += (long)bz * dBS;

  const int m0   = blockIdx.x * 128;
  const int n0   = blockIdx.y * 64;
  const int tid  = threadIdx.x;
  const int wave = tid >> 5;
  const int lane = tid & 31;
  const int lr   = lane & 15;        // M (A frag) / N (B frag) within tile
  const int kh   = (lane >> 4) << 1; // K pair select: 0 or 2
  const int wm   = (wave & 3) * 32;  // wave M offset within block tile
  const int wn   = (wave >> 2) * 32; // wave N offset within block tile

  v8f acc[4];                        // [mi*2+ni] 2x2 subtiles
#pragma unroll
  for (int t = 0; t < 4; ++t) acc[t] = (v8f)0.0f;

  for (int kc = 0; kc < K; kc += 32) {
    // stage A: 128x32 (1024 float4 / 256 threads)
#pragma unroll
    for (int i = 0; i < 4; ++i) {
      int idx = tid + i * 256;
      int r = idx >> 3;
      int c = (idx & 7) << 2;
      *(float4*)&As[r][c] = *(const float4*)(A + (long)(m0 + r) * lda + (kc + c));
    }
    // stage B transposed: 32x64 in, [n][k] out (512 float4 / 256 threads)
#pragma unroll
    for (int i = 0; i < 2; ++i) {
      int idx = tid + i * 256;
      int r = idx >> 4;          // k row
      int c = (idx & 15) << 2;   // n col
      float4 v = *(const float4*)(Bw + (long)(kc + r) * ldb + (n0 + c));
      Bt[c + 0][r] = v.x;
      Bt[c + 1][r] = v.y;
      Bt[c + 2][r] = v.z;
      Bt[c + 3][r] = v.w;
    }
    // prefetch next K-chunk while this one computes (global_prefetch_b8)
    if (kc + 32 < K) {
#pragma unroll
      for (int i = 0; i < 4; ++i) {
        int idx = tid + i * 256;
        __builtin_prefetch(A + (long)(m0 + (idx >> 3)) * lda + (kc + 32 + ((idx & 7) << 2)), 0, 1);
      }
#pragma unroll
      for (int i = 0; i < 2; ++i) {
        int idx = tid + i * 256;
        __builtin_prefetch(Bw + (long)(kc + 32 + (idx >> 4)) * ldb + (n0 + ((idx & 15) << 2)), 0, 1);
      }
    }
    __syncthreads();
    const int ar = wm + lr;
    const int br = wn + lr;
#pragma unroll
    for (int kk = 0; kk < 32; kk += 4) {
      v2f a0 = *(const v2f*)&As[ar][kk + kh];        // (M=lr,    K pair)
      v2f a1 = *(const v2f*)&As[ar + 16][kk + kh];   // (M=lr+16, K pair)
      v2f b0 = *(const v2f*)&Bt[br][kk + kh];        // (N=lr,    K pair)
      v2f b1 = *(const v2f*)&Bt[br + 16][kk + kh];   // (N=lr+16, K pair)
      acc[0] = wmma4(a0, b0, acc[0]);
      acc[1] = wmma4(a0, b1, acc[1]);
      acc[2] = wmma4(a1, b0, acc[2]);
      acc[3] = wmma4(a1, b1, acc[3]);
    }
    __syncthreads();
  }

  // C/D layout: VGPR j -> M = j + 8*(lane>=16); N = lane&15
#pragma unroll
  for (int mi = 0; mi < 2; ++mi) {
    const int rb = m0 + wm + mi * 16 + ((lane >> 4) << 3);
#pragma unroll
    for (int ni = 0; ni < 2; ++ni) {
      const int col = n0 + wn + ni * 16 + lr;
      const float bv = bias ? bias[col] : 0.0f;
      const v8f a = acc[mi * 2 + ni];
#pragma unroll
      for (int j = 0; j < 8; ++j) {
        const int row = rb + j;
        float v = a[j];
        if (mode == 1) {
          v = Xin[(long)row * ldx + col] + fmaxf(v / degs[row] + bv, 0.0f);
        } else {
          v += bv;
        }
        C[(long)row * ldc + col] = v;
      }
    }
  }
}

// ---------------------------------------------------------------------------
// Scores: for a 128(n) x 32(m) block of one batch b, compute Q_e . K_e^T for
// all 3 edges and both heads (12 accumulators / wave: 2 heads x 3 edges x
// 2 m-tiles), then per-element select by adj (adj picks exactly one edge),
// mask zeros to NEG, and write directly into d_out's graph_attention region
// in (H,B,N,N) order. K staged in 16-wide chunks to fit LDS (38.4 KB).
// ---------------------------------------------------------------------------
__global__ __launch_bounds__(256) void scores_wmma(
    const float* __restrict__ Q,   // (E,B,N,D)
    const float* __restrict__ Km,  // (E,B,N,D)
    const int*   __restrict__ adj, // (B,N,N)
    float* __restrict__ gatt) {    // (H,B,N,N)
  __shared__ float Qs[E_][128][20];  // [e][n][k], k-chunk 16 (+4 pad)
  __shared__ float Ks[E_][32][20];   // [e][m][k]

  const int n0  = blockIdx.x * 128;
  const int m0  = blockIdx.y * 32;
  const int b   = blockIdx.z;
  const int tid = threadIdx.x;
  const int wave = tid >> 5;
  const int lane = tid & 31;
  const int lr   = lane & 15;
  const int kh   = (lane >> 4) << 1;

  v8f a0[E_][2], a1[E_][2];  // head 0 (cols 0..127) / head 1 (cols 128..255)
#pragma unroll
  for (int e = 0; e < E_; ++e)
#pragma unroll
    for (int mt = 0; mt < 2; ++mt) { a0[e][mt] = (v8f)0.0f; a1[e][mt] = (v8f)0.0f; }

  for (int kc = 0; kc < D_; kc += 16) {
    // stage Q tiles: per edge 128x16 (512 float4 each)
    for (int e = 0; e < E_; ++e) {
      const float* Qe = Q + ((long)e * B_ + b) * ND_;
#pragma unroll
      for (int i = 0; i < 2; ++i) {
        int idx = tid + i * 256;
        int r = idx >> 2;
        int c = (idx & 3) << 2;
        *(float4*)&Qs[e][r][c] = *(const float4*)(Qe + (long)(n0 + r) * D_ + (kc + c));
      }
    }
    // stage K tiles: per edge 32x16 (384 float4 total)
#pragma unroll
    for (int i = 0; i < 2; ++i) {
      int idx = tid + i * 256;
      if (idx < E_ * 128) {
        int e = idx >> 7;
        int rem = idx & 127;
        int r = rem >> 2;
        int c = (rem & 3) << 2;
        const float* Ke = Km + ((long)e * B_ + b) * ND_;
        *(float4*)&Ks[e][r][c] = *(const float4*)(Ke + (long)(m0 + r) * D_ + (kc + c));
      }
    }
    __syncthreads();
    const int ar = wave * 16 + lr;
    if (kc < 128) {  // head 0
#pragma unroll
      for (int kk = 0; kk < 16; kk += 4) {
#pragma unroll
        for (int e = 0; e < E_; ++e) {
          v2f qf = *(const v2f*)&Qs[e][ar][kk + kh];
          v2f k0 = *(const v2f*)&Ks[e][lr][kk + kh];
          v2f k1 = *(const v2f*)&Ks[e][lr + 16][kk + kh];
          a0[e][0] = wmma4(qf, k0, a0[e][0]);
          a0[e][1] = wmma4(qf, k1, a0[e][1]);
        }
      }
    } else {         // head 1
#pragma unroll
      for (int kk = 0; kk < 16; kk += 4) {
#pragma unroll
        for (int e = 0; e < E_; ++e) {
          v2f qf = *(const v2f*)&Qs[e][ar][kk + kh];
          v2f k0 = *(const v2f*)&Ks[e][lr][kk + kh];
          v2f k1 = *(const v2f*)&Ks[e][lr + 16][kk + kh];
          a1[e][0] = wmma4(qf, k0, a1[e][0]);
          a1[e][1] = wmma4(qf, k1, a1[e][1]);
        }
      }
    }
    __syncthreads();
  }

  const float scale = 0.08838834764831845f;  // 1/sqrt(dk), dk=128
  const int nb = n0 + wave * 16 + ((lane >> 4) << 3);
#pragma unroll
  for (int mt = 0; mt < 2; ++mt) {
    const int m = m0 + mt * 16 + lr;
#pragma unroll
    for (int j = 0; j < 8; ++j) {
      const int n = nb + j;
      const int e = adj[((long)b * N_ + n) * N_ + m];
      float s0, s1;
      if (e == 0) {
        s0 = NEGV; s1 = NEGV;
      } else {
        float r0 = (e == 1) ? a0[0][mt][j] : ((e == 2) ? a0[1][mt][j] : a0[2][mt][j]);
        float r1 = (e == 1) ? a1[0][mt][j] : ((e == 2) ? a1[1][mt][j] : a1[2][mt][j]);
        s0 = r0 * scale;
        s1 = r1 * scale;
        if (s0 == 0.0f) s0 = NEGV;  // reference: scores==0 -> NEG
        if (s1 == 0.0f) s1 = NEGV;
      }
      gatt[((long)(0 * B_ + b) * N_ + n) * N_ + m] = s0;
      gatt[((long)(1 * B_ + b) * N_ + n) * N_ + m] = s1;
    }
  }
}

// ---------------------------------------------------------------------------
// Row softmax over last dim (2048) in place; also emits degs[row] = sum of the
// normalized row (matches reference's numeric re-sum; ~1.0).
// One 256-thread block per row, 8 elems/thread.
// ---------------------------------------------------------------------------
__global__ __launch_bounds__(256) void softmax_rows(
    float* __restrict__ gatt, float* __restrict__ degs) {
  __shared__ float red[256];
  const long row = blockIdx.x;  // over H*B*N
  float* p = gatt + row * N_;
  const int tid = threadIdx.x;

  float4 a = *(float4*)(p + tid * 8);
  float4 c = *(float4*)(p + tid * 8 + 4);
  float x[8] = {a.x, a.y, a.z, a.w, c.x, c.y, c.z, c.w};

  float mx = x[0];
#pragma unroll
  for (int i = 1; i < 8; ++i) mx = fmaxf(mx, x[i]);
  red[tid] = mx; __syncthreads();
  for (int s = 128; s > 0; s >>= 1) {
    if (tid < s) red[tid] = fmaxf(red[tid], red[tid + s]);
    __syncthreads();
  }
  mx = red[0]; __syncthreads();

  float lsum = 0.0f;
#pragma unroll
  for (int i = 0; i < 8; ++i) { x[i] = __expf(x[i] - mx); lsum += x[i]; }
  red[tid] = lsum; __syncthreads();
  for (int s = 128; s > 0; s >>= 1) {
    if (tid < s) red[tid] += red[tid + s];
    __syncthreads();
  }
  const float inv = 1.0f / red[0]; __syncthreads();

  float dsum = 0.0f;
#pragma unroll
  for (int i = 0; i < 8; ++i) { x[i] *= inv; dsum += x[i]; }
  float4 o0 = {x[0], x[1], x[2], x[3]};
  float4 o1 = {x[4], x[5], x[6], x[7]};
  *(float4*)(p + tid * 8)     = o0;
  *(float4*)(p + tid * 8 + 4) = o1;

  red[tid] = dsum; __syncthreads();
  for (int s = 128; s > 0; s >>= 1) {
    if (tid < s) red[tid] += red[tid + s];
    __syncthreads();
  }
  if (tid == 0) degs[row] = red[0];
}

// ---------------------------------------------------------------------------
extern "C" void kernel_launch(void* const* d_in, const int* in_sizes, int n_in,
                              void* d_out, int out_size, void* d_ws, size_t ws_size,
                              hipStream_t stream) {
  (void)in_sizes; (void)n_in; (void)out_size; (void)ws_size;
  const float* X   = (const float*)d_in[0];  // (B,N,D)
  const int*   adj = (const int*)  d_in[1];  // (B,N,N)
  const float* Wq  = (const float*)d_in[2];  // (E,D,D)
  const float* bq  = (const float*)d_in[3];  // (E,D)
  const float* Wk  = (const float*)d_in[4];  // (E,D,D)
  const float* bk  = (const float*)d_in[5];  // (E,D)
  const float* gW  = (const float*)d_in[6];  // (I,D,D)
  const float* gb  = (const float*)d_in[7];  // (I,D)
  const float* aW  = (const float*)d_in[8];  // (2D,D)
  const float* ab  = (const float*)d_in[9];  // (D)

  float* out  = (float*)d_out;               // (B,N,D)
  float* gatt = out + (long)BN_ * D_;        // (H,B,N,N) — reference's transposed attn

  float* ws   = (float*)d_ws;
  float* Qb   = ws;                           // (E,B,N,D)
  float* Kb   = Qb  + (long)E_ * BN_ * D_;    // (E,B,N,D)
  float* h1   = Kb  + (long)E_ * BN_ * D_;    // (B,N,D) scratch, reused
  float* x0   = h1  + (long)BN_ * D_;         // (B,N,D) head-0 state
  float* x1   = x0  + (long)BN_ * D_;         // (B,N,D) head-1 state
  float* cat  = x1  + (long)BN_ * D_;         // (B,N,2D)
  float* degs = cat + (long)BN_ * 2 * D_;     // (H,B,N)

  const dim3 blk(256);
  const dim3 gFull(BN_ / 128, D_ / 64, 1);    // M=8192, Nc=256
  const dim3 gBat(N_ / 128, D_ / 64, B_);     // M=2048, Nc=256, batched over b

  // 1) Q/K projections: Q_e = X@Wq[e]+bq[e], K_e = X@Wk[e]+bk[e]
  for (int e = 0; e < E_; ++e) {
    gemm_wmma_f32<<<gFull, blk, 0, stream>>>(
        X, D_, 0, Wq + (long)e * D_ * D_, D_, 0, bq + (long)e * D_,
        Qb + (long)e * BN_ * D_, D_, 0, nullptr, 0, 0, nullptr, 0, D_, 0);
    gemm_wmma_f32<<<gFull, blk, 0, stream>>>(
        X, D_, 0, Wk + (long)e * D_ * D_, D_, 0, bk + (long)e * D_,
        Kb + (long)e * BN_ * D_, D_, 0, nullptr, 0, 0, nullptr, 0, D_, 0);
  }

  // 2) scores (edge-select + zero mask) -> gatt, then 3) softmax + degs
  scores_wmma<<<dim3(N_ / 128, N_ / 32, B_), blk, 0, stream>>>(Qb, Kb, adj, gatt);
  softmax_rows<<<dim3(H_ * B_ * N_), blk, 0, stream>>>(gatt, degs);

  // 4) GCN iter 0: h1 = X@gcn_W[0] (shared across heads)
  gemm_wmma_f32<<<gFull, blk, 0, stream>>>(
      X, D_, 0, gW, D_, 0, nullptr, h1, D_, 0, nullptr, 0, 0, nullptr, 0, D_, 0);

  // 5) x_h = X + relu(attn_h @ h1 / degs_h + gcn_b[0])   (fused epilogue)
  for (int h = 0; h < H_; ++h) {
    gemm_wmma_f32<<<gBat, blk, 0, stream>>>(
        gatt + (long)h * B_ * NN_, N_, NN_,   // A = attn_h (b-batched), K=2048
        h1, D_, ND_,                          // B = h1[b]
        gb,                                   // bias = gcn_b[0]
        (h ? x1 : x0), D_, ND_,               // C = x_h
        X, D_, ND_,                           // Xin = nodes_embed
        degs + (long)h * B_ * N_, N_,         // degs_h[b]
        N_, 1);
  }

  // 6) GCN iter 1 per head: h1 = x_h@gcn_W[1]; cat[:,:,h*D:] = x_h + relu(...)
  for (int h = 0; h < H_; ++h) {
    float* xh = h ? x1 : x0;
    gemm_wmma_f32<<<gFull, blk, 0, stream>>>(
        xh, D_, 0, gW + (long)D_ * D_, D_, 0, nullptr,
        h1, D_, 0, nullptr, 0, 0, nullptr, 0, D_, 0);
    gemm_wmma_f32<<<gBat, blk, 0, stream>>>(
        gatt + (long)h * B_ * NN_, N_, NN_,
        h1, D_, ND_,
        gb + D_,                              // gcn_b[1]
        cat + (long)h * D_, 2 * D_, (long)N_ * 2 * D_,
        xh, D_, ND_,
        degs + (long)h * B_ * N_, N_,
        N_, 1);
  }

  // 7) out = cat @ agg_W + agg_b   (K = 512)
  gemm_wmma_f32<<<gFull, blk, 0, stream>>>(
      cat, 2 * D_, 0, aW, D_, 0, ab,
      out, D_, 0, nullptr, 0, 0, nullptr, 0, 2 * D_, 0);
}